// Aggregator_32564442038933
// MI455X (gfx1250) — compile-verified
//
#include <hip/hip_runtime.h>

typedef float v2f __attribute__((ext_vector_type(2)));
typedef float v8f __attribute__((ext_vector_type(8)));

#define BLOCK_PAIRS 16   // (b,e) pairs per block; 16-aligned => single b per block

__global__ __launch_bounds__(256, 4)
void kgcn_aggregate_kernel(const float* __restrict__ self_vectors,     // [B*E, 64]
                           const float* __restrict__ neighbor_vectors, // [B*E, 32, 64]
                           const float* __restrict__ neighbor_relations,// [B*E, 32, 64]
                           const float* __restrict__ user_embeddings,  // [B, 64]
                           const float* __restrict__ W,                // [64, 128]
                           const float* __restrict__ bias,             // [64]
                           float* __restrict__ out)                    // [B*E, 64]
{
    __shared__ float uLds[64];
    __shared__ float X[BLOCK_PAIRS][130];   // padded row stride: avoids LDS bank conflicts

    const int tid      = threadIdx.x;
    const int lane     = tid & 31;
    const int wave     = tid >> 5;
    const int pairBase = blockIdx.x * BLOCK_PAIRS;
    const int bIdx     = pairBase >> 6;     // uniform: 16-aligned pair range stays in one b

    // stage u[b] (64 floats) into LDS once per block
    if (tid < 32) {
        const float2 u2 = *(const float2*)(user_embeddings + (size_t)bIdx * 64 + 2 * tid);
        uLds[2 * tid]     = u2.x;
        uLds[2 * tid + 1] = u2.y;
    }
    __syncthreads();

    // ---------------- Phase A: scores -> softmax -> weighted aggregation ----------------
    #pragma unroll
    for (int rep = 0; rep < 2; ++rep) {
        const int lp   = wave + rep * 8;            // local pair 0..15
        const int pair = pairBase + lp;
        const float* rel = neighbor_relations + ((size_t)pair * 32 + lane) * 64;
        const float* nvb = neighbor_vectors  + (size_t)pair * 2048;

        // warm the neighbor_vectors tile while we do the relation dots
        __builtin_prefetch(nvb + (size_t)lane * 64, 0, 1);
        __builtin_prefetch(nvb + (size_t)lane * 64 + 32, 0, 1);

        // lane = neighbor: dot(u, rel[n,:]) over D=64 with float4 loads (256B/lane contiguous)
        float acc = 0.f;
        #pragma unroll
        for (int i = 0; i < 16; ++i) {
            const float4 r4 = *(const float4*)(rel + i * 4);
            const float4 u4 = *(const float4*)(uLds + i * 4);
            acc = fmaf(r4.x, u4.x, acc);
            acc = fmaf(r4.y, u4.y, acc);
            acc = fmaf(r4.z, u4.z, acc);
            acc = fmaf(r4.w, u4.w, acc);
        }

        // wave32 softmax over the 32 neighbors
        float mx = acc;
        #pragma unroll
        for (int off = 16; off > 0; off >>= 1)
            mx = fmaxf(mx, __shfl_xor(mx, off, 32));
        const float ex = __expf(acc - mx);
        float sum = ex;
        #pragma unroll
        for (int off = 16; off > 0; off >>= 1)
            sum += __shfl_xor(sum, off, 32);
        const float pn = ex / sum;                  // p[lane]

        // lane = feature pair (d = 2*lane, 2*lane+1): coalesced float2 loads over n
        float a0 = 0.f, a1 = 0.f;
        #pragma unroll
        for (int n = 0; n < 32; ++n) {
            const float  pv = __shfl(pn, n, 32);    // broadcast p[n]
            const float2 v  = *(const float2*)(nvb + n * 64 + 2 * lane);
            a0 = fmaf(pv, v.x, a0);
            a1 = fmaf(pv, v.y, a1);
        }

        // x = concat(self, agg) staged to LDS for the WMMA GEMM
        const float2 sv = *(const float2*)(self_vectors + (size_t)pair * 64 + 2 * lane);
        X[lp][2 * lane]          = sv.x;
        X[lp][2 * lane + 1]      = sv.y;
        X[lp][64 + 2 * lane]     = a0;
        X[lp][64 + 2 * lane + 1] = a1;
    }
    __syncthreads();

    // ---------------- Phase B: out[16x64] = X[16x128] @ W^T[128x64], WMMA f32 ----------------
    if (wave < 4) {
        const int half = lane >> 4;              // 0: K={k0,k0+1}, rows 0-7 ; 1: K={k0+2,k0+3}, rows 8-15
        const int mrow = lane & 15;              // A-matrix row (pair)
        const int nc   = wave * 16 + (lane & 15);// output column of this N-tile
        const float bv = bias[nc];

        v8f c = {};
        #pragma unroll
        for (int k = 0; k < 32; ++k) {
            const int k0 = k * 4 + half * 2;
            v2f a, bb;
            a.x = X[mrow][k0];
            a.y = X[mrow][k0 + 1];
            const float2 w2 = *(const float2*)(W + (size_t)nc * 128 + k0); // B = W^T (K-major per col)
            bb.x = w2.x;
            bb.y = w2.y;
            // D = A(16x4,f32) * B(4x16,f32) + C
            c = __builtin_amdgcn_wmma_f32_16x16x4_f32(
                    false, a, false, bb, (short)0, c, false, false);
        }

        // C/D layout: VGPR r -> (M = r + 8*half, N = lane&15); bias + ReLU epilogue
        #pragma unroll
        for (int r = 0; r < 8; ++r) {
            const int mm = r + half * 8;
            float val = c[r] + bv;
            val = fmaxf(val, 0.f);
            out[(size_t)(pairBase + mm) * 64 + nc] = val;
        }
    }
}

extern "C" void kernel_launch(void* const* d_in, const int* in_sizes, int n_in,
                              void* d_out, int out_size, void* d_ws, size_t ws_size,
                              hipStream_t stream) {
    (void)in_sizes; (void)n_in; (void)out_size; (void)d_ws; (void)ws_size;
    const float* self_vectors       = (const float*)d_in[0];
    const float* neighbor_vectors   = (const float*)d_in[1];
    const float* neighbor_relations = (const float*)d_in[2];
    const float* user_embeddings    = (const float*)d_in[3];
    const float* W                  = (const float*)d_in[4];
    const float* bias               = (const float*)d_in[5];
    float* out                      = (float*)d_out;

    const int total_pairs = 1024 * 64;                 // B * E
    const int grid = total_pairs / BLOCK_PAIRS;        // 4096 blocks
    kgcn_aggregate_kernel<<<grid, 256, 0, stream>>>(
        self_vectors, neighbor_vectors, neighbor_relations,
        user_embeddings, W, bias, out);
}